// Block_34935263986141
// MI455X (gfx1250) — compile-verified
//
#include <hip/hip_runtime.h>
#include <hip/hip_bf16.h>
#include <math.h>

// Problem constants (match reference)
#define NN 100000
#define CC 128
#define FC 512
#define KK 49
#define PP 100000
#define EPSF 1e-6f

typedef __attribute__((ext_vector_type(2))) float v2f;
typedef __attribute__((ext_vector_type(8))) float v8f;

// ---------------- zero-init: y accumulator [N,C] and gx2[512] ----------------
__global__ void zero_kernel(float* __restrict__ y, float* __restrict__ gx2, size_t ny) {
    size_t i = (size_t)blockIdx.x * blockDim.x + threadIdx.x;
    if (i < ny) y[i] = 0.0f;
    if (i < FC) gx2[i] = 0.0f;
}

// ------------- repack W[K,N] -> Wp[q][n] = {W[2q][n], W[2q+1][n]} ------------
// lets each WMMA B-fragment be a single global_load_b64
__global__ void pack_rowpairs(const float* __restrict__ W, float2* __restrict__ Wp,
                              int nrows, int ncols) {
    int i = blockIdx.x * blockDim.x + threadIdx.x;
    int total = (nrows / 2) * ncols;
    if (i >= total) return;
    int q = i / ncols, n = i - q * ncols;
    Wp[i] = make_float2(W[(size_t)(2 * q) * ncols + n],
                        W[(size_t)(2 * q + 1) * ncols + n]);
}

// ---------------- sparse depthwise conv: one wave per (k,p) pair -------------
// y[out_map[k,p], :] += x[in_map[k,p], :] * w_dw[k, :]
__global__ __launch_bounds__(256) void dw_scatter(
    const float* __restrict__ x, const float* __restrict__ w_dw,
    const int* __restrict__ in_map, const int* __restrict__ out_map,
    float* __restrict__ y, int kp) {
    int wave = (int)(((size_t)blockIdx.x * blockDim.x + threadIdx.x) >> 5);
    int lane = threadIdx.x & 31;
    if (wave >= kp) return;
    int pair = __builtin_amdgcn_readfirstlane(wave);       // uniform -> SGPR
    int k  = pair / PP;
    int im = __builtin_amdgcn_readfirstlane(in_map[pair]);
    int om = __builtin_amdgcn_readfirstlane(out_map[pair]);
    const float4* xr = (const float4*)(x + (size_t)im * CC);
    const float4* wr = (const float4*)(w_dw + (size_t)k * CC);
    float4 xv = xr[lane];
    float4 wv = wr[lane];
    float* yr = y + (size_t)om * CC + lane * 4;
    atomicAdd(yr + 0, xv.x * wv.x);
    atomicAdd(yr + 1, xv.y * wv.y);
    atomicAdd(yr + 2, xv.z * wv.z);
    atomicAdd(yr + 3, xv.w * wv.w);
}

// ---------------- GEMM1: LN(y+b_dw) @ W1 + b1 -> GELU -> h; partial gx2 ------
// Block: 16 rows x 512 cols. 8 waves, each owns 64 cols (4 WMMA N-tiles).
#define A1S 132   // padded LDS row stride (floats) to avoid bank conflicts
__global__ __launch_bounds__(256) void gemm1_kernel(
    const float* __restrict__ y, const float* __restrict__ b_dw,
    const float* __restrict__ ln_g, const float* __restrict__ ln_b,
    const float2* __restrict__ W1p, const float* __restrict__ b1,
    float* __restrict__ h, float* __restrict__ gx2) {
    __shared__ float As[16 * A1S];
    const int row0 = blockIdx.x * 16;
    const int tid  = threadIdx.x;
    // cooperative load of 16x128 A tile + depthwise bias
    for (int i = tid; i < 16 * CC; i += 256) {
        int r = i >> 7, c = i & (CC - 1);
        As[r * A1S + c] = y[(size_t)(row0 + r) * CC + c] + b_dw[c];
    }
    __syncthreads();
    const int wave = tid >> 5, lane = tid & 31;
    // LayerNorm: wave w normalizes rows 2w and 2w+1
    for (int rr = 0; rr < 2; ++rr) {
        int r = wave * 2 + rr;
        float s1 = 0.f, s2 = 0.f;
        for (int c = lane; c < CC; c += 32) {
            float v = As[r * A1S + c];
            s1 += v; s2 += v * v;
        }
        for (int off = 16; off >= 1; off >>= 1) {
            s1 += __shfl_xor(s1, off, 32);
            s2 += __shfl_xor(s2, off, 32);
        }
        float mu  = s1 * (1.0f / CC);
        float var = s2 * (1.0f / CC) - mu * mu;
        float rstd = rsqrtf(var + EPSF);
        for (int c = lane; c < CC; c += 32) {
            As[r * A1S + c] = (As[r * A1S + c] - mu) * rstd * ln_g[c] + ln_b[c];
        }
    }
    __syncthreads();
    // fp32 WMMA 16x16x4 main loop
    const int lhalf = lane >> 4;     // 0: K pair {0,1}, 1: K pair {2,3}
    const int l16   = lane & 15;
    const int n0    = wave * 64;
    v8f acc0 = {}, acc1 = {}, acc2 = {}, acc3 = {};
    for (int kk = 0; kk < CC; kk += 4) {
        v2f a;
        a.x = As[l16 * A1S + kk + lhalf * 2 + 0];
        a.y = As[l16 * A1S + kk + lhalf * 2 + 1];
        const float2* wrow = W1p + (size_t)((kk >> 1) + lhalf) * FC;
        float2 t0 = wrow[n0 + 0 * 16 + l16];
        float2 t1 = wrow[n0 + 1 * 16 + l16];
        float2 t2 = wrow[n0 + 2 * 16 + l16];
        float2 t3 = wrow[n0 + 3 * 16 + l16];
        v2f b;
        b.x = t0.x; b.y = t0.y;
        acc0 = __builtin_amdgcn_wmma_f32_16x16x4_f32(false, a, false, b, (short)0, acc0, false, false);
        b.x = t1.x; b.y = t1.y;
        acc1 = __builtin_amdgcn_wmma_f32_16x16x4_f32(false, a, false, b, (short)0, acc1, false, false);
        b.x = t2.x; b.y = t2.y;
        acc2 = __builtin_amdgcn_wmma_f32_16x16x4_f32(false, a, false, b, (short)0, acc2, false, false);
        b.x = t3.x; b.y = t3.y;
        acc3 = __builtin_amdgcn_wmma_f32_16x16x4_f32(false, a, false, b, (short)0, acc3, false, false);
    }
    // epilogue: bias + exact GELU, store h, accumulate per-channel sum of squares
    v8f* accs[4] = { &acc0, &acc1, &acc2, &acc3 };
    #pragma unroll
    for (int t = 0; t < 4; ++t) {
        int n = n0 + t * 16 + l16;
        float bias = b1[n];
        float ss = 0.f;
        #pragma unroll
        for (int i = 0; i < 8; ++i) {
            int m = lhalf * 8 + i;
            float v = (*accs[t])[i] + bias;
            v = 0.5f * v * (1.0f + erff(v * 0.7071067811865475f));  // exact GELU
            h[(size_t)(row0 + m) * FC + n] = v;
            ss += v * v;
        }
        ss += __shfl_xor(ss, 16, 32);       // combine M halves (same column n)
        if (lhalf == 0) atomicAdd(&gx2[n], ss);
    }
}

// ---------------- GRN scalar stage: s[c] = 1 + gamma[c]*nx[c] ----------------
__global__ void grn_scale(const float* __restrict__ gx2,
                          const float* __restrict__ grn_gamma,
                          float* __restrict__ s) {
    __shared__ float red[16];
    int c = threadIdx.x;                 // 512 threads
    float g = sqrtf(gx2[c]);
    float v = g;
    for (int off = 16; off >= 1; off >>= 1) v += __shfl_xor(v, off, 32);
    if ((c & 31) == 0) red[c >> 5] = v;
    __syncthreads();
    if (c < 16) {
        float t = red[c];
        for (int off = 8; off >= 1; off >>= 1) t += __shfl_xor(t, off, 32);
        if (c == 0) red[0] = t;
    }
    __syncthreads();
    float mean = red[0] * (1.0f / FC);
    float nx = g / (mean + EPSF);
    s[c] = 1.0f + grn_gamma[c] * nx;
}

// ---------------- GEMM2: (h*s + beta) @ W2 + b2 + x -> out -------------------
// Block: 16 rows x 128 cols. 8 waves, each owns one 16-col N-tile. K=512.
#define A2S 516   // padded LDS row stride (floats)
__global__ __launch_bounds__(256) void gemm2_kernel(
    const float* __restrict__ h, const float* __restrict__ s,
    const float* __restrict__ grn_beta, const float2* __restrict__ W2p,
    const float* __restrict__ b2, const float* __restrict__ x,
    float* __restrict__ out) {
    __shared__ float As[16 * A2S];
    const int row0 = blockIdx.x * 16;
    const int tid  = threadIdx.x;
    // load 16x512 h tile with GRN scale/bias applied
    for (int i = tid; i < 16 * FC; i += 256) {
        int r = i >> 9, c = i & (FC - 1);
        As[r * A2S + c] = h[(size_t)(row0 + r) * FC + c] * s[c] + grn_beta[c];
    }
    __syncthreads();
    const int wave = tid >> 5, lane = tid & 31;
    const int lhalf = lane >> 4;
    const int l16   = lane & 15;
    const int n0    = wave * 16;
    v8f acc = {};
    for (int kk = 0; kk < FC; kk += 4) {
        v2f a, b;
        a.x = As[l16 * A2S + kk + lhalf * 2 + 0];
        a.y = As[l16 * A2S + kk + lhalf * 2 + 1];
        float2 t = W2p[(size_t)((kk >> 1) + lhalf) * CC + n0 + l16];
        b.x = t.x; b.y = t.y;
        acc = __builtin_amdgcn_wmma_f32_16x16x4_f32(false, a, false, b, (short)0, acc, false, false);
    }
    int n = n0 + l16;
    float bias = b2[n];
    #pragma unroll
    for (int i = 0; i < 8; ++i) {
        int m = lhalf * 8 + i;
        size_t idx = (size_t)(row0 + m) * CC + n;
        out[idx] = acc[i] + bias + x[idx];
    }
}

// ---------------- launcher ----------------
extern "C" void kernel_launch(void* const* d_in, const int* in_sizes, int n_in,
                              void* d_out, int out_size, void* d_ws, size_t ws_size,
                              hipStream_t stream) {
    const float* x      = (const float*)d_in[0];
    const float* w_dw   = (const float*)d_in[1];
    const float* b_dw   = (const float*)d_in[2];
    const float* ln_g   = (const float*)d_in[3];
    const float* ln_b   = (const float*)d_in[4];
    const float* W1     = (const float*)d_in[5];
    const float* b1     = (const float*)d_in[6];
    const float* grn_g  = (const float*)d_in[7];
    const float* grn_b  = (const float*)d_in[8];
    const float* W2     = (const float*)d_in[9];
    const float* b2     = (const float*)d_in[10];
    const int*   in_map  = (const int*)d_in[11];
    const int*   out_map = (const int*)d_in[12];
    float* out = (float*)d_out;

    // workspace layout: y [N,C] | h [N,4C] | gx2 [512] | s [512] | W1p | W2p
    float* ws   = (float*)d_ws;
    float* y    = ws;
    float* h    = y + (size_t)NN * CC;
    float* gx2  = h + (size_t)NN * FC;
    float* sfac = gx2 + FC;
    float2* W1p = (float2*)(sfac + FC);            // (CC/2) x FC pairs
    float2* W2p = W1p + (size_t)(CC / 2) * FC;     // (FC/2) x CC pairs

    size_t ny = (size_t)NN * CC;
    zero_kernel<<<(unsigned)((ny + 255) / 256), 256, 0, stream>>>(y, gx2, ny);

    // one-time per launch weight repacks (amortized over 6250 tile blocks)
    pack_rowpairs<<<((CC / 2) * FC + 255) / 256, 256, 0, stream>>>(W1, W1p, CC, FC);
    pack_rowpairs<<<((FC / 2) * CC + 255) / 256, 256, 0, stream>>>(W2, W2p, FC, CC);

    int kp = KK * PP;                              // 4.9M pairs, 1 wave each
    dw_scatter<<<(unsigned)((kp + 7) / 8), 256, 0, stream>>>(x, w_dw, in_map, out_map, y, kp);

    gemm1_kernel<<<NN / 16, 256, 0, stream>>>(y, b_dw, ln_g, ln_b, W1p, b1, h, gx2);

    grn_scale<<<1, FC, 0, stream>>>(gx2, grn_g, sfac);

    gemm2_kernel<<<NN / 16, 256, 0, stream>>>(h, sfac, grn_b, W2p, b2, x, out);
}